// Model_22016002359933
// MI455X (gfx1250) — compile-verified
//
#include <hip/hip_runtime.h>
#include <math.h>

#define BQ 8
#define LQ 512
#define DM 256
#define DI 512
#define DSn 16
#define DTR 16
#define DFF 1024
#define PREDQ 32

typedef float v2f __attribute__((ext_vector_type(2)));
typedef float v8f __attribute__((ext_vector_type(8)));

__device__ __forceinline__ float apply_act(float v, int act) {
    if (act == 1) return fmaxf(v, 0.0f);
    if (act == 2) return (v > 20.0f) ? v : log1pf(expf(v));
    return v;
}

// ---------------------------------------------------------------------------
// Generic fp32 WMMA GEMM (1 wave = 16x16 tile): C = act(X[M,K](lda)*W[N,K]^T + bias)
// act: 0=none, 1=relu, 2=softplus.  Used for odd shapes (N=48, M=8).
// ---------------------------------------------------------------------------
__global__ void gemm_f32_wmma(const float* __restrict__ X, const float* __restrict__ W,
                              const float* __restrict__ bias, float* __restrict__ C,
                              int M, int N, int K, int lda, int ldc, int act)
{
    int wave = blockIdx.x * (blockDim.x >> 5) + (threadIdx.x >> 5);
    int lane = threadIdx.x & 31;
    int tilesN = N >> 4;
    int tilesM = (M + 15) >> 4;
    if (wave >= tilesM * tilesN) return;           // wave-uniform: EXEC all-1s at WMMA
    int tm = wave / tilesN, tn = wave - tm * tilesN;
    int m0 = tm << 4, n0 = tn << 4;

    int arow = m0 + (lane & 15); if (arow >= M) arow = M - 1;   // clamped; rows>=M never stored
    int col  = n0 + (lane & 15);
    int khalf = (lane >> 4) << 1;                  // lanes 0-15: K=k,k+1 ; lanes 16-31: K=k+2,k+3
    const float* xr = X + (size_t)arow * lda + khalf;
    const float* wr = W + (size_t)col  * K   + khalf;

    v8f acc = {};
    for (int k = 0; k < K; k += 4) {
        v2f a, b;
        a.x = xr[k]; a.y = xr[k + 1];
        b.x = wr[k]; b.y = wr[k + 1];
        acc = __builtin_amdgcn_wmma_f32_16x16x4_f32(false, a, false, b, (short)0, acc,
                                                    false, false);
    }

    float bv = bias ? bias[col] : 0.0f;
    int rbase = m0 + ((lane >> 4) << 3);           // C/D: VGPR v -> rows m0+v / m0+v+8
#pragma unroll
    for (int v = 0; v < 8; ++v) {
        int r = rbase + v;
        if (r < M) C[(size_t)r * ldc + col] = apply_act(acc[v] + bv, act);
    }
}

// ---------------------------------------------------------------------------
// Register-blocked fp32 WMMA GEMM (1 wave = 32x32 tile, 2x2 fragments).
// 4 independent accumulators hide WMMA RAW latency; A/B fragments reused 2x,
// so 4 b64 loads feed 4 WMMAs per k-step.  Requires M%32==0 && N%32==0.
// ---------------------------------------------------------------------------
__global__ void gemm_f32_wmma_2x2(const float* __restrict__ X, const float* __restrict__ W,
                                  const float* __restrict__ bias, float* __restrict__ C,
                                  int M, int N, int K, int lda, int ldc, int act)
{
    int wave = blockIdx.x * (blockDim.x >> 5) + (threadIdx.x >> 5);
    int lane = threadIdx.x & 31;
    int tilesN = N >> 5;
    int tilesM = M >> 5;
    if (wave >= tilesM * tilesN) return;           // wave-uniform exit
    int tm = wave / tilesN, tn = wave - tm * tilesN;
    int m0 = tm << 5, n0 = tn << 5;

    int khalf = (lane >> 4) << 1;
    const float* xr0 = X + (size_t)(m0 + (lane & 15)) * lda + khalf;
    const float* xr1 = xr0 + (size_t)16 * lda;
    const float* wr0 = W + (size_t)(n0 + (lane & 15)) * K + khalf;
    const float* wr1 = wr0 + (size_t)16 * K;

    v8f a00 = {}, a01 = {}, a10 = {}, a11 = {};
    for (int k = 0; k < K; k += 4) {
        v2f A0, A1, B0, B1;
        A0.x = xr0[k]; A0.y = xr0[k + 1];
        A1.x = xr1[k]; A1.y = xr1[k + 1];
        B0.x = wr0[k]; B0.y = wr0[k + 1];
        B1.x = wr1[k]; B1.y = wr1[k + 1];
        a00 = __builtin_amdgcn_wmma_f32_16x16x4_f32(false, A0, false, B0, (short)0, a00, false, false);
        a01 = __builtin_amdgcn_wmma_f32_16x16x4_f32(false, A0, false, B1, (short)0, a01, false, false);
        a10 = __builtin_amdgcn_wmma_f32_16x16x4_f32(false, A1, false, B0, (short)0, a10, false, false);
        a11 = __builtin_amdgcn_wmma_f32_16x16x4_f32(false, A1, false, B1, (short)0, a11, false, false);
    }

    int rbase = m0 + ((lane >> 4) << 3);
    int c0 = n0 + (lane & 15), c1 = c0 + 16;
    float bv0 = bias ? bias[c0] : 0.0f;
    float bv1 = bias ? bias[c1] : 0.0f;
#pragma unroll
    for (int v = 0; v < 8; ++v) {
        size_t r0 = (size_t)(rbase + v) * ldc;
        size_t r1 = (size_t)(rbase + v + 16) * ldc;
        C[r0 + c0] = apply_act(a00[v] + bv0, act);
        C[r0 + c1] = apply_act(a01[v] + bv1, act);
        C[r1 + c0] = apply_act(a10[v] + bv0, act);
        C[r1 + c1] = apply_act(a11[v] + bv1, act);
    }
}

// ---------------------------------------------------------------------------
// Elementwise / normalization / scan kernels
// ---------------------------------------------------------------------------
__global__ void k_add_pos(float* x, const float* __restrict__ pos, int n) {
    int i = blockIdx.x * blockDim.x + threadIdx.x;
    if (i < n) x[i] += pos[i % (LQ * DM)];
}

__global__ void k_flip(const float* __restrict__ x, float* __restrict__ xf) {
    int i = blockIdx.x * blockDim.x + threadIdx.x;
    if (i >= BQ * LQ * DM) return;
    int d = i % DM; int t = (i / DM) % LQ; int b = i / (DM * LQ);
    xf[i] = x[((size_t)b * LQ + (LQ - 1 - t)) * DM + d];
}

__global__ void k_add_flip(float* __restrict__ nx, const float* __restrict__ r) {
    int i = blockIdx.x * blockDim.x + threadIdx.x;
    if (i >= BQ * LQ * DM) return;
    int d = i % DM; int t = (i / DM) % LQ; int b = i / (DM * LQ);
    nx[i] += r[((size_t)b * LQ + (LQ - 1 - t)) * DM + d];
}

// instance norm over time axis; saves mean/std for final de-normalization
__global__ void k_instnorm(float* x, float* means, float* stdev) {
    int i = blockIdx.x * blockDim.x + threadIdx.x;
    if (i >= BQ * DM) return;
    int d = i % DM; int b = i / DM;
    float* p = x + (size_t)b * LQ * DM + d;
    float s = 0.0f;
    for (int t = 0; t < LQ; ++t) s += p[(size_t)t * DM];
    float mean = s * (1.0f / LQ);
    float vs = 0.0f;
    for (int t = 0; t < LQ; ++t) { float dv = p[(size_t)t * DM] - mean; vs += dv * dv; }
    float sd = sqrtf(vs * (1.0f / LQ) + 1e-8f);
    means[i] = mean; stdev[i] = sd;
    float inv = 1.0f / sd;
    for (int t = 0; t < LQ; ++t) p[(size_t)t * DM] = (p[(size_t)t * DM] - mean) * inv;
}

// depthwise causal conv (k=4) + bias + SiLU ; xin = xz[..., 0:DI]
__global__ void k_conv_silu(const float* __restrict__ xz, const float* __restrict__ cw,
                            const float* __restrict__ cb, float* __restrict__ u) {
    int i = blockIdx.x * blockDim.x + threadIdx.x;
    if (i >= BQ * LQ * DI) return;
    int d = i % DI; int t = (i / DI) % LQ; int b = i / (DI * LQ);
    const float* base = xz + (size_t)b * LQ * (2 * DI) + d;
    float s = cb[d];
#pragma unroll
    for (int j = 0; j < 4; ++j) {
        int tt = t - 3 + j;
        if (tt >= 0) s += cw[d * 4 + j] * base[(size_t)tt * (2 * DI)];
    }
    u[i] = s / (1.0f + expf(-s));                  // silu
}

// selective scan: one lane per (b,d,n); 16-lane shfl_xor reduction for y = h.C
// fused: y = (sum_n h*C + u*D) * silu(z)
__global__ void k_scan(const float* __restrict__ delta, const float* __restrict__ u,
                       const float* __restrict__ xdbl, const float* __restrict__ A_log,
                       const float* __restrict__ Dp, const float* __restrict__ xz,
                       float* __restrict__ y) {
    int idx = blockIdx.x * blockDim.x + threadIdx.x;   // BQ*DI*16 lanes
    int n = idx & 15;
    int d = (idx >> 4) & (DI - 1);
    int b = idx >> 13;
    float A  = -expf(A_log[d * DSn + n]);
    float Dd = Dp[d];
    const float* xb = xdbl + (size_t)b * LQ * 48;
    const float* dl = delta + (size_t)b * LQ * DI + d;
    const float* uu = u     + (size_t)b * LQ * DI + d;
    const float* zz = xz + (size_t)b * LQ * (2 * DI) + DI + d;
    float* yo = y + (size_t)b * LQ * DI + d;
    float h = 0.0f;
    for (int t = 0; t < LQ; ++t) {
        float dt = dl[(size_t)t * DI];
        float ut = uu[(size_t)t * DI];
        float Bv = xb[(size_t)t * 48 + DTR + n];
        float Cv = xb[(size_t)t * 48 + DTR + DSn + n];
        h = expf(dt * A) * h + (dt * ut) * Bv;
        float prod = h * Cv;
        prod += __shfl_xor(prod, 1, 32);
        prod += __shfl_xor(prod, 2, 32);
        prod += __shfl_xor(prod, 4, 32);
        prod += __shfl_xor(prod, 8, 32);
        if (n == 0) {
            float zt = zz[(size_t)t * (2 * DI)];
            float sz = zt / (1.0f + expf(-zt));
            yo[(size_t)t * DI] = (prod + ut * Dd) * sz;
        }
    }
}

// wave-per-row LayerNorm of (xin [+ addv]) over DM=256; in-place safe
__global__ void k_add_ln(const float* __restrict__ xin, const float* __restrict__ addv,
                         const float* __restrict__ g, const float* __restrict__ bb,
                         float* __restrict__ out) {
    int wave = blockIdx.x * (blockDim.x >> 5) + (threadIdx.x >> 5);
    int lane = threadIdx.x & 31;
    if (wave >= BQ * LQ) return;
    const float* xr = xin + (size_t)wave * DM;
    float vals[8]; float s = 0.0f;
#pragma unroll
    for (int i = 0; i < 8; ++i) {
        float v = xr[lane + i * 32];
        if (addv) v += addv[(size_t)wave * DM + lane + i * 32];
        vals[i] = v; s += v;
    }
#pragma unroll
    for (int o = 16; o > 0; o >>= 1) s += __shfl_xor(s, o, 32);
    float mean = s * (1.0f / DM);
    float vs = 0.0f;
#pragma unroll
    for (int i = 0; i < 8; ++i) { float dv = vals[i] - mean; vs += dv * dv; }
#pragma unroll
    for (int o = 16; o > 0; o >>= 1) vs += __shfl_xor(vs, o, 32);
    float rstd = rsqrtf(vs * (1.0f / DM) + 1e-5f);
    float* orow = out + (size_t)wave * DM;
#pragma unroll
    for (int i = 0; i < 8; ++i) {
        int c = lane + i * 32;
        orow[c] = (vals[i] - mean) * rstd * g[c] + bb[c];
    }
}

__global__ void k_globalfeat(const float* __restrict__ x, float* __restrict__ gf) {
    int i = blockIdx.x * blockDim.x + threadIdx.x;
    if (i >= BQ * DM) return;
    int d = i % DM, b = i / DM;
    const float* p = x + (size_t)b * LQ * DM + d;
    float s = 0.0f, mx = -3.4e38f;
    for (int t = 0; t < LQ; ++t) { float v = p[(size_t)t * DM]; s += v; mx = fmaxf(mx, v); }
    gf[b * 768 + d]          = s * (1.0f / LQ);
    gf[b * 768 + DM + d]     = mx;
    gf[b * 768 + 2 * DM + d] = p[(size_t)(LQ - 1) * DM];
}

__global__ void k_pool3(const float* __restrict__ gf, float* __restrict__ gfp) {
    int i = blockIdx.x * blockDim.x + threadIdx.x;
    if (i >= BQ * DM) return;
    int d = i % DM, b = i / DM;
    const float* g = gf + b * 768 + d * 3;
    gfp[i] = (g[0] + g[1] + g[2]) * (1.0f / 3.0f);
}

__global__ void k_final(const float* __restrict__ fo, const float* __restrict__ stdev,
                        const float* __restrict__ means, float* __restrict__ out) {
    int i = blockIdx.x * blockDim.x + threadIdx.x;
    if (i >= BQ * PREDQ * DM) return;
    int d = i % DM; int b = i / (PREDQ * DM);
    out[i] = fo[i] * stdev[b * DM + d] + means[b * DM + d];
}

// ---------------------------------------------------------------------------
// Host launcher
// ---------------------------------------------------------------------------
static inline void gemm(const float* X, const float* W, const float* bias, float* C,
                        int M, int N, int K, int lda, int ldc, int act, hipStream_t s) {
    if ((M % 32 == 0) && (N % 32 == 0)) {
        int tiles = (M / 32) * (N / 32);
        int blocks = (tiles + 3) / 4;               // 4 waves (128 threads) per block
        gemm_f32_wmma_2x2<<<blocks, 128, 0, s>>>(X, W, bias, C, M, N, K, lda, ldc, act);
    } else {
        int tiles = ((M + 15) / 16) * (N / 16);
        int blocks = (tiles + 3) / 4;
        gemm_f32_wmma<<<blocks, 128, 0, s>>>(X, W, bias, C, M, N, K, lda, ldc, act);
    }
}

extern "C" void kernel_launch(void* const* d_in, const int* in_sizes, int n_in,
                              void* d_out, int out_size, void* d_ws, size_t ws_size,
                              hipStream_t stream)
{
    (void)out_size; (void)ws_size;
    // ---- resolve input ordering (top level: x_enc first vs params first) ----
    int ix_xenc, pbase;
    if (in_sizes[0] == BQ * LQ * DM) { ix_xenc = 0; pbase = 2; }
    else                             { pbase = 0; ix_xenc = n_in - 1; }
    auto P = [&](int i) -> const float* { return (const float*)d_in[pbase + i]; };

    // ---- resolve params flatten order: jax-sorted keys vs insertion order ----
    bool sorted = (in_sizes[pbase] != DM * DM);     // sorted leaf0 = enc_ln_b [256]; insertion = in_w [65536]
    int I_enc_ln_g, I_enc_ln_b, I_fin_w, I_fin_b, I_in_w, I_in_b;
    int I_out1_w, I_out1_b, I_out2_w, I_out2_b, I_pos;
    int L0; const int Lstride = 26;
    int R_b1, R_b2, R_ln1g, R_ln1b, R_ln2g, R_ln2b, R_mf, R_mr, R_w1, R_w2;
    int M_A, M_D, M_cb, M_cw, M_dtb, M_dtw, M_in, M_out, M_xp;
    if (sorted) {
        I_enc_ln_b = 0; I_enc_ln_g = 1; I_fin_b = 2; I_fin_w = 3; I_in_b = 4; I_in_w = 5;
        L0 = 6;
        I_out1_b = 84; I_out1_w = 85; I_out2_b = 86; I_out2_w = 87; I_pos = 88;
        R_b1 = 0; R_b2 = 1; R_ln1b = 2; R_ln1g = 3; R_ln2b = 4; R_ln2g = 5;
        R_mf = 6; R_mr = 15; R_w1 = 24; R_w2 = 25;
        M_A = 0; M_D = 1; M_cb = 2; M_cw = 3; M_dtb = 4; M_dtw = 5; M_in = 6; M_out = 7; M_xp = 8;
    } else {
        I_in_w = 0; I_in_b = 1; I_pos = 2; /* time_w=3, time_b=4 */
        L0 = 5;
        I_enc_ln_g = 83; I_enc_ln_b = 84; I_out1_w = 85; I_out1_b = 86;
        I_out2_w = 87; I_out2_b = 88; I_fin_w = 89; I_fin_b = 90;
        R_mf = 0; R_mr = 9; R_w1 = 18; R_b1 = 19; R_w2 = 20; R_b2 = 21;
        R_ln1g = 22; R_ln1b = 23; R_ln2g = 24; R_ln2b = 25;
        M_in = 0; M_cw = 1; M_cb = 2; M_xp = 3; M_dtw = 4; M_dtb = 5; M_A = 6; M_D = 7; M_out = 8;
    }

    // ---- workspace carve-up (floats), ~60 MB total ----
    float* ws    = (float*)d_ws;
    float* x     = ws;                // [B,L,DM]   1048576
    float* xf    = ws + 1048576;      // flipped input
    float* newx  = ws + 2097152;      // mamba sum
    float* tmp1  = ws + 3145728;      // reverse-dir out / FFN out
    float* bufA  = ws + 4194304;      // xz [B,L,2*DI] or FFN hidden (4194304)
    float* ub    = ws + 8388608;      // u  [B,L,DI]   (2097152)
    float* dl    = ws + 10485760;     // delta         (2097152)
    float* ysc   = ws + 12582912;     // scan output   (2097152)
    float* xdbl  = ws + 14680064;     // [B,L,48]      (196608)
    float* means = ws + 14876672;     // [B,DM]
    float* stdev = ws + 14878720;     // [B,DM]
    float* gf    = ws + 14880768;     // [B,768]
    float* gfp   = ws + 14886912;     // [B,DM]
    float* h1    = ws + 14888960;     // [B,DFF]
    float* o2    = ws + 14897152;     // [B,PRED*DM]
    float* fo    = ws + 14962688;     // [B*PRED,DM]

    const int EW = 256;
    const int nTot = BQ * LQ * DM;
    const int M4 = BQ * LQ;           // 4096 rows

    // embed: x = x_enc @ in_w.T + in_b + pos ; then instance norm over time
    gemm((const float*)d_in[ix_xenc], P(I_in_w), P(I_in_b), x, M4, DM, DM, DM, DM, 0, stream);
    k_add_pos<<<(nTot + EW - 1) / EW, EW, 0, stream>>>(x, P(I_pos), nTot);
    k_instnorm<<<(BQ * DM + EW - 1) / EW, EW, 0, stream>>>(x, means, stdev);

    for (int l = 0; l < 3; ++l) {
        int lb = L0 + l * Lstride;
        k_flip<<<(nTot + EW - 1) / EW, EW, 0, stream>>>(x, xf);
        for (int dir = 0; dir < 2; ++dir) {
            int mb = lb + (dir ? R_mr : R_mf);
            const float* xin = dir ? xf : x;
            gemm(xin, P(mb + M_in), nullptr, bufA, M4, 2 * DI, DM, DM, 2 * DI, 0, stream);     // in_proj
            k_conv_silu<<<(BQ * LQ * DI + EW - 1) / EW, EW, 0, stream>>>(bufA, P(mb + M_cw),
                                                                         P(mb + M_cb), ub);
            gemm(ub, P(mb + M_xp), nullptr, xdbl, M4, 48, DI, DI, 48, 0, stream);              // x_proj
            gemm(xdbl, P(mb + M_dtw), P(mb + M_dtb), dl, M4, DI, DTR, 48, DI, 2, stream);      // dt_proj+softplus
            k_scan<<<(BQ * DI * 16) / EW, EW, 0, stream>>>(dl, ub, xdbl, P(mb + M_A),
                                                           P(mb + M_D), bufA, ysc);
            gemm(ysc, P(mb + M_out), nullptr, dir ? tmp1 : newx, M4, DM, DI, DI, DM, 0, stream); // out_proj
        }
        k_add_flip<<<(nTot + EW - 1) / EW, EW, 0, stream>>>(newx, tmp1);
        k_add_ln<<<(M4 + 7) / 8, 256, 0, stream>>>(x, newx, P(lb + R_ln1g), P(lb + R_ln1b), x);
        gemm(x, P(lb + R_w1), P(lb + R_b1), bufA, M4, DFF, DM, DM, DFF, 1, stream);            // FFN up+relu
        gemm(bufA, P(lb + R_w2), P(lb + R_b2), tmp1, M4, DM, DFF, DFF, DM, 0, stream);         // FFN down
        k_add_ln<<<(M4 + 7) / 8, 256, 0, stream>>>(x, tmp1, P(lb + R_ln2g), P(lb + R_ln2b), x);
    }

    k_add_ln<<<(M4 + 7) / 8, 256, 0, stream>>>(x, nullptr, P(I_enc_ln_g), P(I_enc_ln_b), x);
    k_globalfeat<<<(BQ * DM + EW - 1) / EW, EW, 0, stream>>>(x, gf);
    k_pool3<<<(BQ * DM + EW - 1) / EW, EW, 0, stream>>>(gf, gfp);
    gemm(gfp, P(I_out1_w), P(I_out1_b), h1, BQ, DFF, DM, DM, DFF, 1, stream);
    gemm(h1, P(I_out2_w), P(I_out2_b), o2, BQ, PREDQ * DM, DFF, DFF, PREDQ * DM, 0, stream);
    gemm(o2, P(I_fin_w), P(I_fin_b), fo, BQ * PREDQ, DM, DM, DM, DM, 0, stream);
    k_final<<<(BQ * PREDQ * DM + EW - 1) / EW, EW, 0, stream>>>(fo, stdev, means, (float*)d_out);
}